// LKBallWholePoseScoringModule_60902636257309
// MI455X (gfx1250) — compile-verified
//
#include <hip/hip_runtime.h>
#include <hip/hip_bf16.h>

// ---- problem constants (match reference) ----
constexpr int kP   = 2;
constexpr int kB   = 96;
constexpr int kA   = 32;
constexpr int kN   = kB * kA;          // 3072 atoms per pose
constexpr int kMD  = 4;
constexpr int kMA  = 4;
constexpr int kNPOLB = kMD + kMA;      // 8 polar sites per block
constexpr int kNPOL  = kB * kNPOLB;    // 768 polar sites per pose
constexpr int kNS    = kP * kNPOL;     // 1536 sites total
constexpr float kCLK = 0.0897935610625833f;
constexpr float kLOG2E = 1.44269504088896f;

constexpr int kNTI = kNPOL / 16;       // 48 i-tiles per pose
constexpr int kNJT = kN / 16;          // 192 j-tiles per pose
constexpr int kWAVES = 8;              // waves per workgroup (256 threads, wave32)

typedef float v2f __attribute__((ext_vector_type(2)));
typedef float v8f __attribute__((ext_vector_type(8)));

// D(16x16,f32) = A(16x4,f32) * B(4x16,f32) ; K=4 (zero-padded xyz)
__device__ inline v8f wmma_dot(v2f a, v2f b) {
  v8f c = {};
  return __builtin_amdgcn_wmma_f32_16x16x4_f32(
      /*neg_a=*/false, a, /*neg_b=*/false, b,
      /*c_mod=*/(short)0, c, /*reuse_a=*/false, /*reuse_b=*/false);
}

// shared per-tile element math (8 C-rows per lane).
// NOTE: the cutoff mask selects the CHEAP coefficient (cndmask), never the
// expensive exp/rsq result -- keeps the compiler from if-converting the
// element body into exec-masked skip blocks that serialize the unroll.
__device__ inline void tile_accum(v8f dotP, v8f dotW0, v8f dotW1,
                                  float rj2, float rj, float voljm,
                                  const float* r_ri, const float* r_invl,
                                  const float* r_ri2, const float* r_w02,
                                  const float* r_w12, const float* cw,
                                  float cutoff, float wtop, float invramp,
                                  float& s0, float& s1)
{
#pragma unroll
  for (int k = 0; k < 8; ++k) {
    float d2 = fmaxf(r_ri2[k] + rj2 - 2.0f*dotP[k], 0.01f);
    float rr    = __builtin_amdgcn_rsqf(d2);     // v_rsq_f32
    float d     = d2 * rr;
    float invd2 = rr * rr;
    float xx = (d - (r_ri[k] + rj)) * r_invl[k];
    float ex = __builtin_amdgcn_exp2f(xx * xx * (-kLOG2E));  // v_exp_f32
    float cc = (d < cutoff) ? cw[k] : 0.0f;      // single cndmask
    float lk = cc * voljm * ex * invd2;          // ex/invd2 always live
    float dw0 = r_w02[k] + rj2 - 2.0f*dotW0[k];
    float dw1 = r_w12[k] + rj2 - 2.0f*dotW1[k];
    float dmn = fminf(dw0, dw1);
    float wt  = fminf(fmaxf((wtop - dmn) * invramp, 0.0f), 1.0f);
    float frac = wt * wt * (3.0f - 2.0f*wt);
    s0 += lk;
    s1 += lk * frac;
  }
}

// ---------------- Phase 1a: build waters + per-site params ----------------
__global__ __launch_bounds__(256)
void lkball_precompute(const float* __restrict__ coords,     // (P,N,3)
                       const int*   __restrict__ block_type, // (P,B)
                       const int*   __restrict__ bonds,      // (NT,A,2)
                       const int*   __restrict__ ranges,     // (NT,A,2)
                       const int*   __restrict__ n_donH,     // (NT,)
                       const int*   __restrict__ n_acc,      // (NT,)
                       const int*   __restrict__ donH_inds,  // (NT,MD)
                       const int*   __restrict__ don_hvy,    // (NT,MD)
                       const int*   __restrict__ acc_inds,   // (NT,MA)
                       const int*   __restrict__ hyb,        // (NT,MA)
                       const float* __restrict__ lkb_params, // (NT,A,4)
                       const float* __restrict__ wgen,       // (2,)
                       const float* __restrict__ sp2t,
                       const float* __restrict__ sp3t,
                       const float* __restrict__ ringt,
                       float* __restrict__ wsf,
                       int*   __restrict__ wsi,
                       float* __restrict__ out)
{
  int idx = blockIdx.x * blockDim.x + threadIdx.x;
  if (idx < 2 * kP) out[idx] = 0.0f;           // zero the 4 output scalars
  if (idx >= kNS) return;

  int p   = idx / kNPOL;
  int rem = idx - p * kNPOL;
  int b   = rem / kNPOLB;
  int s   = rem - b * kNPOLB;
  int bt  = block_type[p * kB + b];

  const float wdist = wgen[0];
  const float wang  = wgen[1];
  const float* cb = coords + ((size_t)p * kN + (size_t)b * kA) * 3;

  float w0x=0.f, w0y=0.f, w0z=0.f, w1x=0.f, w1y=0.f, w1z=0.f;
  bool m0, m1, pmask;
  int loc;

  if (s < kMD) {
    // donor: water along heavy->H direction
    loc = don_hvy[bt * kMD + s];
    int Hi = donH_inds[bt * kMD + s];
    float hx = cb[loc*3+0], hy = cb[loc*3+1], hz = cb[loc*3+2];
    float dx = cb[Hi*3+0]-hx, dy = cb[Hi*3+1]-hy, dz = cb[Hi*3+2]-hz;
    float inv = rsqrtf(dx*dx + dy*dy + dz*dz + 1e-12f);
    w0x = hx + wdist*dx*inv; w0y = hy + wdist*dy*inv; w0z = hz + wdist*dz*inv;
    pmask = (s < n_donH[bt]);
    m0 = pmask; m1 = false;
  } else {
    // acceptor: two waters from local frame (e1, e2, nrm) and torsion table
    int a = s - kMD;
    loc = acc_inds[bt * kMA + a];
    int r1     = ranges[(bt*kA + loc)   * 2 + 0];
    int basei  = bonds [(bt*kA + r1)    * 2 + 1];
    int r2     = ranges[(bt*kA + basei) * 2 + 0];
    int base2i = bonds [(bt*kA + r2)    * 2 + 1];

    float cx = cb[loc*3+0],    cy = cb[loc*3+1],    cz = cb[loc*3+2];
    float bx = cb[basei*3+0],  by = cb[basei*3+1],  bz = cb[basei*3+2];
    float ax = cb[base2i*3+0], ay = cb[base2i*3+1], az = cb[base2i*3+2];

    float e1x = cx-bx, e1y = cy-by, e1z = cz-bz;
    float inv = rsqrtf(e1x*e1x + e1y*e1y + e1z*e1z + 1e-12f);
    e1x *= inv; e1y *= inv; e1z *= inv;

    float vx = bx-ax, vy = by-ay, vz = bz-az;
    float nx = vy*e1z - vz*e1y;
    float ny = vz*e1x - vx*e1z;
    float nz = vx*e1y - vy*e1x;
    inv = rsqrtf(nx*nx + ny*ny + nz*nz + 1e-12f);
    nx *= inv; ny *= inv; nz *= inv;

    float e2x = ny*e1z - nz*e1y;
    float e2y = nz*e1x - nx*e1z;
    float e2z = nx*e1y - ny*e1x;

    int h = hyb[bt * kMA + a];
    const float* tors = (h == 0) ? sp2t : ((h == 1) ? sp3t : ringt);
    float ct = cosf(wang), st = sinf(wang);
    float bxx = cx - wdist*ct*e1x;
    float bxy = cy - wdist*ct*e1y;
    float bxz = cz - wdist*ct*e1z;

    float c0 = cosf(tors[0]), s0v = sinf(tors[0]);
    float c1 = cosf(tors[1]), s1v = sinf(tors[1]);
    w0x = bxx + wdist*st*(c0*e2x + s0v*nx);
    w0y = bxy + wdist*st*(c0*e2y + s0v*ny);
    w0z = bxz + wdist*st*(c0*e2z + s0v*nz);
    w1x = bxx + wdist*st*(c1*e2x + s1v*nx);
    w1y = bxy + wdist*st*(c1*e2y + s1v*ny);
    w1z = bxz + wdist*st*(c1*e2z + s1v*nz);

    pmask = (a < n_acc[bt]);
    m0 = pmask; m1 = pmask;
  }

  // masked waters -> far sentinel so smoothstep weight clamps to 0
  const float SENT = 1.0e4f;
  if (!m0) { w0x = w0y = w0z = SENT; }
  if (!m1) { w1x = w1y = w1z = SENT; }

  float px = cb[loc*3+0], py = cb[loc*3+1], pz = cb[loc*3+2];
  const float* par = lkb_params + (size_t)(bt*kA + loc) * 4;
  float ri = par[0], dgi = par[1], lami = par[2];

  wsf[ 0*kNS + idx] = px;
  wsf[ 1*kNS + idx] = py;
  wsf[ 2*kNS + idx] = pz;
  wsf[ 3*kNS + idx] = ri;
  wsf[ 4*kNS + idx] = pmask ? (kCLK * dgi / lami) : 0.0f;  // pol_mask folded in
  wsf[ 5*kNS + idx] = 1.0f / lami;
  wsf[ 6*kNS + idx] = px*px + py*py + pz*pz;
  wsf[ 7*kNS + idx] = w0x;
  wsf[ 8*kNS + idx] = w0y;
  wsf[ 9*kNS + idx] = w0z;
  wsf[10*kNS + idx] = w1x;
  wsf[11*kNS + idx] = w1y;
  wsf[12*kNS + idx] = w1z;
  wsf[13*kNS + idx] = w0x*w0x + w0y*w0y + w0z*w0z;
  wsf[14*kNS + idx] = w1x*w1x + w1y*w1y + w1z*w1z;
  wsi[idx] = loc;
}

// ---------------- Phase 1b: pack per-atom j-side params ----------------
__global__ __launch_bounds__(256)
void lkball_jparams(const int*   __restrict__ block_type,
                    const float* __restrict__ lkb_params,
                    const int*   __restrict__ is_h,
                    float* __restrict__ jp)
{
  int idx = blockIdx.x * blockDim.x + threadIdx.x;
  if (idx >= kP * kN) return;
  int p = idx / kN;
  int j = idx - p * kN;
  int bj = j >> 5, aj = j & 31;
  int bt = block_type[p * kB + bj];
  const float* par = lkb_params + (size_t)(bt * kA + aj) * 4;
  jp[idx * 2 + 0] = par[0];
  jp[idx * 2 + 1] = (is_h[bt * kA + aj] == 0) ? par[3] : 0.0f;
}

// ---------------- Phase 2: WMMA-tiled pairwise scoring ----------------
__global__ __launch_bounds__(256)
void lkball_main(const float* __restrict__ coords,
                 const int*   __restrict__ block_type,
                 const int*   __restrict__ minsep,     // (P,B,B)
                 const int*   __restrict__ path,       // (NT,A,A)
                 const float* __restrict__ lkg,        // (3,)
                 const float* __restrict__ wsf,
                 const int*   __restrict__ wsi,
                 const float* __restrict__ jp,         // (P,N,2)
                 float* __restrict__ out)
{
  const int tile = blockIdx.x;          // p * kNTI + it
  const int p    = tile / kNTI;
  const int it   = tile % kNTI;
  const int tid  = threadIdx.x;
  const int lane = tid & 31;
  // wave index forced into an SGPR -> scalar loop control & branches
  const int wv   = __builtin_amdgcn_readfirstlane(tid >> 5);
  const int hi   = lane >> 4;           // high half of wave (truly divergent)
  const int nl   = lane & 15;

  const float cutoff  = lkg[0];
  const float ramp2   = lkg[1];
  const float d2low   = lkg[2];
  const float invramp = 1.0f / ramp2;
  const float wtop    = d2low + ramp2;   // (wtop - d2wmin)/ramp2 then clamp

  const float* X    = wsf +  0*kNS;
  const float* Y    = wsf +  1*kNS;
  const float* Z    = wsf +  2*kNS;
  const float* RI   = wsf +  3*kNS;
  const float* COEF = wsf +  4*kNS;
  const float* INVL = wsf +  5*kNS;
  const float* RI2  = wsf +  6*kNS;
  const float* W0X  = wsf +  7*kNS;
  const float* W0Y  = wsf +  8*kNS;
  const float* W0Z  = wsf +  9*kNS;
  const float* W1X  = wsf + 10*kNS;
  const float* W1Y  = wsf + 11*kNS;
  const float* W1Z  = wsf + 12*kNS;
  const float* W02  = wsf + 13*kNS;
  const float* W12  = wsf + 14*kNS;

  // --- A matrices (16x4, K = xyz0). Lanes 0-15 carry K=0,1 ; lanes 16-31 K=2,3.
  const int mrow = p*kNPOL + it*16 + nl;    // A-row M = nl for both lane halves
  v2f a_pol, a_w0, a_w1;
  {
    float x = X[mrow],  y = Y[mrow],  z = Z[mrow];
    a_pol[0] = hi ? z : x;  a_pol[1] = hi ? 0.0f : y;
    x = W0X[mrow]; y = W0Y[mrow]; z = W0Z[mrow];
    a_w0[0]  = hi ? z : x;  a_w0[1]  = hi ? 0.0f : y;
    x = W1X[mrow]; y = W1Y[mrow]; z = W1Z[mrow];
    a_w1[0]  = hi ? z : x;  a_w1[1]  = hi ? 0.0f : y;
  }

  // --- per-element row data: this lane's C rows are m = 8*hi + k (k=0..7),
  //     all inside the same block b_i = it*2 + hi.
  float r_ri[8], r_coef[8], r_invl[8], r_ri2[8], r_w02[8], r_w12[8];
  int   r_ai[8];
  const int rbase = p*kNPOL + it*16 + 8*hi;
#pragma unroll
  for (int k = 0; k < 8; ++k) {
    int r = rbase + k;
    r_ri[k]   = RI[r];   r_coef[k] = COEF[r]; r_invl[k] = INVL[r];
    r_ri2[k]  = RI2[r];  r_w02[k]  = W02[r];  r_w12[k]  = W12[r];
    r_ai[k]   = wsi[r];
  }
  const int  b_i   = it*2 + hi;
  const int  bt_i  = block_type[p*kB + b_i];
  const int* pathI = path + (size_t)bt_i * kA * kA;
  // per-lane-half minsep row (one b32 load per tile after pointer select)
  const int* msrow = minsep + ((size_t)p*kB + b_i) * kB;
  const float* jprow = jp + (size_t)p * kN * 2;

  // The intra-block (path-distance) case hits bj in {it*2, it*2+1}, i.e.
  // jt in [4it, 4it+3].  This wave only sees jt == wv (mod 8): AT MOST ONE
  // such tile.  Peel it out of the main loop entirely.
  const int r4  = (wv - 4*it) & 7;
  const int tjt = (r4 < 4) ? (4*it + r4) : -1;   // scalar

  float s0 = 0.0f, s1 = 0.0f;

  // ---- main loop: branch-free body, inter-block separation only ----
  for (int jt = wv; jt < kNJT; jt += kWAVES) {
    if (jt == tjt) continue;                       // scalar cmp + branch
    const int bj = jt >> 1;
    const int aj = ((jt & 1) << 4) | nl;
    const int j  = (bj << 5) | aj;

    const float* cj = coords + ((size_t)p*kN + j) * 3;
    float xj = cj[0], yj = cj[1], zj = cj[2];
    float rj   = jprow[j*2 + 0];
    float volj = jprow[j*2 + 1];                   // 0 for hydrogens

    v2f bm;
    bm[0] = hi ? zj : xj;
    bm[1] = hi ? 0.0f : yj;

    v8f dotP  = wmma_dot(a_pol, bm);
    v8f dotW0 = wmma_dot(a_w0, bm);
    v8f dotW1 = wmma_dot(a_w1, bm);

    float rj2 = xj*xj + yj*yj + zj*zj;
    int   msep  = msrow[bj];
    float voljm = (msep >= 4) ? volj : 0.0f;       // sep mask folded: 1 cndmask

    tile_accum(dotP, dotW0, dotW1, rj2, rj, voljm,
               r_ri, r_invl, r_ri2, r_w02, r_w12, r_coef,
               cutoff, wtop, invramp, s0, s1);
  }

  // ---- peeled tile: intra-block path distances (runs at most once) ----
  if (tjt >= 0) {
    const int jt = tjt;
    const int bj = jt >> 1;
    const int aj = ((jt & 1) << 4) | nl;
    const int j  = (bj << 5) | aj;

    const float* cj = coords + ((size_t)p*kN + j) * 3;
    float xj = cj[0], yj = cj[1], zj = cj[2];
    float rj   = jprow[j*2 + 0];
    float volj = jprow[j*2 + 1];

    v2f bm;
    bm[0] = hi ? zj : xj;
    bm[1] = hi ? 0.0f : yj;

    v8f dotP  = wmma_dot(a_pol, bm);
    v8f dotW0 = wmma_dot(a_w0, bm);
    v8f dotW1 = wmma_dot(a_w1, bm);

    float rj2 = xj*xj + yj*yj + zj*zj;
    int msep = msrow[bj];

    float cv[8];
#pragma unroll
    for (int k = 0; k < 8; ++k) {
      int pd  = pathI[r_ai[k]*kA + aj];
      int sep = (bj == b_i) ? pd : msep;           // hi-half may be inter-block
      cv[k] = (sep >= 4) ? r_coef[k] : 0.0f;
    }

    tile_accum(dotP, dotW0, dotW1, rj2, rj, volj,
               r_ri, r_invl, r_ri2, r_w02, r_w12, cv,
               cutoff, wtop, invramp, s0, s1);
  }

  // wave32 reduction, then one atomic per wave per output
#pragma unroll
  for (int off = 16; off > 0; off >>= 1) {
    s0 += __shfl_down(s0, off, 32);
    s1 += __shfl_down(s1, off, 32);
  }
  if (lane == 0) {
    atomicAdd(out + p,       s0);
    atomicAdd(out + kP + p,  s1);
  }
}

// ---------------- host launcher ----------------
extern "C" void kernel_launch(void* const* d_in, const int* in_sizes, int n_in,
                              void* d_out, int out_size, void* d_ws, size_t ws_size,
                              hipStream_t stream) {
  const float* pose_coords = (const float*)d_in[0];
  const int*   block_type  = (const int*)  d_in[1];
  const int*   minsep      = (const int*)  d_in[2];
  // d_in[3] bt_n_atoms unused (constant A)
  const int*   bonds       = (const int*)  d_in[4];
  const int*   ranges      = (const int*)  d_in[5];
  const int*   n_donH      = (const int*)  d_in[6];
  const int*   n_acc       = (const int*)  d_in[7];
  const int*   donH_inds   = (const int*)  d_in[8];
  const int*   don_hvy     = (const int*)  d_in[9];
  const int*   acc_inds    = (const int*)  d_in[10];
  const int*   hyb         = (const int*)  d_in[11];
  const int*   is_h        = (const int*)  d_in[12];
  const float* lkb_params  = (const float*)d_in[13];
  const int*   path        = (const int*)  d_in[14];
  const float* lkg         = (const float*)d_in[15];
  const float* wgen        = (const float*)d_in[16];
  const float* sp2t        = (const float*)d_in[17];
  const float* sp3t        = (const float*)d_in[18];
  const float* ringt       = (const float*)d_in[19];
  (void)in_sizes; (void)n_in; (void)out_size; (void)ws_size;

  float* out = (float*)d_out;
  float* wsf = (float*)d_ws;
  int*   wsi = (int*)(wsf + 15 * kNS);
  float* jp  = wsf + 16 * kNS;          // (P*N) float2 region

  lkball_precompute<<<(kNS + 255) / 256, 256, 0, stream>>>(
      pose_coords, block_type, bonds, ranges, n_donH, n_acc,
      donH_inds, don_hvy, acc_inds, hyb, lkb_params,
      wgen, sp2t, sp3t, ringt, wsf, wsi, out);

  lkball_jparams<<<(kP * kN + 255) / 256, 256, 0, stream>>>(
      block_type, lkb_params, is_h, jp);

  lkball_main<<<kP * kNTI, 256, 0, stream>>>(
      pose_coords, block_type, minsep, path, lkg, wsf, wsi, jp, out);
}